// Conjugate_Gradient_69879117906154
// MI455X (gfx1250) — compile-verified
//
#include <hip/hip_runtime.h>
#include <hip/hip_bf16.h>
#include <stdint.h>

// ---------------------------------------------------------------------------
// CG solve, n = 8192, 20 iterations.  Memory-bound fp32 GEMV dominates:
//   - p (32KB) staged into LDS per block via gfx1250 async-to-LDS loads
//     (ASYNCcnt-tracked, s_wait_asynccnt)
//   - M rows < NT_SPLIT use RT loads (kept L2-resident across the 20
//     iterations: 160MB of the 192MB global L2), rows >= NT_SPLIT use NT
//     loads so the streaming tail never evicts the resident head.
// No matrix-matrix op exists in this workload (GEMV, N=1), so WMMA would be
// 16x redundant FLOPs on a bandwidth-bound roofline; the async-to-LDS path
// is the CDNA5 feature that actually pays here.
// ---------------------------------------------------------------------------

#define CG_N 8192
#define CG_N4 (CG_N / 4)          // 2048 float4 per row / for p
#define NT_SPLIT 5120             // 5120 rows * 32KB = 160MB kept hot in L2
#define NITER 20

typedef float vf4 __attribute__((ext_vector_type(4)));

// ------------------------- matvec: Ap = M @ p ------------------------------
// First function in the module so the disasm snippet shows the async ops.
// 256 threads = 8 waves; each wave computes 2 consecutive rows (reuses the
// LDS read of p for both rows).  Grid = 8192/16 = 512 blocks.
__global__ __launch_bounds__(256) void cg_matvec(const float* __restrict__ M,
                                                 const float* __restrict__ p,
                                                 float* __restrict__ Ap) {
    __shared__ vf4 s_p[CG_N4];    // 32 KB of the WGP's 320 KB LDS

    const int t = threadIdx.x;
    const vf4* __restrict__ p4 = (const vf4*)p;

    // ---- stage p into LDS with gfx1250 async loads (ASYNCcnt-tracked) ----
    // 2048 float4 / 256 threads = 8 x b128 per thread; each lane moves 16B
    // straight from global to its LDS slot without touching VGPRs.
#pragma unroll
    for (int j = 0; j < 8; ++j) {
        int idx = t + j * 256;
        uint32_t ldsb = (uint32_t)(uintptr_t)(&s_p[idx]);   // low 32b = LDS offset
        const vf4* gsrc = p4 + idx;
        asm volatile("global_load_async_to_lds_b128 %0, %1, off"
                     :: "v"(ldsb), "v"(gsrc)
                     : "memory");
    }
    asm volatile("s_wait_asynccnt 0" ::: "memory");
    __syncthreads();

    const int wid  = t >> 5;
    const int lane = t & 31;
    const int row0 = blockIdx.x * 16 + wid * 2;   // even: both rows same side of split
    const int row1 = row0 + 1;

    const vf4* __restrict__ M0 = (const vf4*)(M + (size_t)row0 * CG_N);
    const vf4* __restrict__ M1 = (const vf4*)(M + (size_t)row1 * CG_N);

    vf4 a0 = (vf4)0.0f, a1 = (vf4)0.0f;

    if (row0 >= NT_SPLIT) {
        // streaming tail: non-temporal so the L2-resident head survives
#pragma unroll 4
        for (int i = lane; i < CG_N4; i += 32) {
            vf4 pv = s_p[i];
            vf4 m0 = __builtin_nontemporal_load(&M0[i]);
            vf4 m1 = __builtin_nontemporal_load(&M1[i]);
            a0 += m0 * pv;
            a1 += m1 * pv;
        }
    } else {
        // hot head: default temporal hint -> stays resident in 192MB L2
#pragma unroll 4
        for (int i = lane; i < CG_N4; i += 32) {
            vf4 pv = s_p[i];
            vf4 m0 = M0[i];
            vf4 m1 = M1[i];
            a0 += m0 * pv;
            a1 += m1 * pv;
        }
    }

    float s0 = a0.x + a0.y + a0.z + a0.w;
    float s1 = a1.x + a1.y + a1.z + a1.w;
    // wave32 butterfly reduction (deterministic order)
#pragma unroll
    for (int off = 16; off > 0; off >>= 1) {
        s0 += __shfl_xor(s0, off, 32);
        s1 += __shfl_xor(s1, off, 32);
    }
    if (lane == 0) {
        Ap[row0] = s0;
        Ap[row1] = s1;
    }
}

// ---------------------------- init: X=0, r=p=RHS ---------------------------
__global__ void cg_init(const float* __restrict__ rhs,
                        float* __restrict__ X,
                        float* __restrict__ r,
                        float* __restrict__ p) {
    int i = blockIdx.x * blockDim.x + threadIdx.x;
    if (i < CG_N) {
        float v = rhs[i];
        X[i] = 0.0f;
        r[i] = v;
        p[i] = v;
    }
}

// ------------------- update: dots, alpha/beta, axpy, freeze ----------------
// Single block of 1024 threads (32 waves); each thread owns 8 elements
// (2 x float4).  Faithful to the reference: active = (sum r^2 > 1e-10);
// when inactive the state is frozen (and since frozen state implies
// unchanged r, recomputing rtr is exactly the carried rTr).
__global__ __launch_bounds__(1024) void cg_update(float* __restrict__ X,
                                                  float* __restrict__ r,
                                                  float* __restrict__ p,
                                                  const float* __restrict__ Ap) {
    __shared__ float s_a[1024];
    __shared__ float s_b[1024];

    const int t = threadIdx.x;

    vf4* __restrict__ X4        = (vf4*)X;
    vf4* __restrict__ r4        = (vf4*)r;
    vf4* __restrict__ p4        = (vf4*)p;
    const vf4* __restrict__ Ap4 = (const vf4*)Ap;

    vf4 rv[2], pv[2], apv[2];
    float rtr_p = 0.0f, pap_p = 0.0f;
#pragma unroll
    for (int j = 0; j < 2; ++j) {
        int i = t + j * 1024;            // float4 index, [0, 2048)
        rv[j]  = r4[i];
        pv[j]  = p4[i];
        apv[j] = Ap4[i];
        vf4 rr = rv[j] * rv[j];
        vf4 pa = pv[j] * apv[j];
        rtr_p += rr.x + rr.y + rr.z + rr.w;
        pap_p += pa.x + pa.y + pa.z + pa.w;
    }
    s_a[t] = rtr_p;
    s_b[t] = pap_p;
    __syncthreads();
#pragma unroll
    for (int s = 512; s > 0; s >>= 1) {
        if (t < s) {
            s_a[t] += s_a[t + s];
            s_b[t] += s_b[t + s];
        }
        __syncthreads();
    }
    const float rtr = s_a[0];
    const float pap = s_b[0];

    if (!(rtr > 1e-10f)) return;          // frozen: uniform across the block

    const float alpha = rtr / pap;

    float rn_p = 0.0f;
#pragma unroll
    for (int j = 0; j < 2; ++j) {
        int i = t + j * 1024;
        X4[i] = X4[i] + alpha * pv[j];
        rv[j] = rv[j] - alpha * apv[j];
        r4[i] = rv[j];
        vf4 rr = rv[j] * rv[j];
        rn_p += rr.x + rr.y + rr.z + rr.w;
    }
    __syncthreads();                      // everyone done reading s_a[0]/s_b[0]
    s_a[t] = rn_p;
    __syncthreads();
#pragma unroll
    for (int s = 512; s > 0; s >>= 1) {
        if (t < s) s_a[t] += s_a[t + s];
        __syncthreads();
    }
    const float beta = s_a[0] / rtr;

#pragma unroll
    for (int j = 0; j < 2; ++j) {
        int i = t + j * 1024;
        p4[i] = rv[j] + beta * pv[j];
    }
}

// ---------------------------------------------------------------------------
extern "C" void kernel_launch(void* const* d_in, const int* in_sizes, int n_in,
                              void* d_out, int out_size, void* d_ws, size_t ws_size,
                              hipStream_t stream) {
    // inputs (setup_inputs order): X [8192] (shape only), M [8192*8192], RHS [8192]
    const float* M   = (const float*)d_in[1];
    const float* RHS = (const float*)d_in[2];

    float* X  = (float*)d_out;            // X lives in the output buffer
    float* ws = (float*)d_ws;
    float* r  = ws;                       //  8192 floats
    float* p  = ws + CG_N;                //  8192 floats
    float* Ap = ws + 2 * CG_N;            //  8192 floats  (96 KB total)

    cg_init<<<CG_N / 256, 256, 0, stream>>>(RHS, X, r, p);

    for (int it = 0; it < NITER; ++it) {
        cg_matvec<<<CG_N / 16, 256, 0, stream>>>(M, p, Ap);
        cg_update<<<1, 1024, 0, stream>>>(X, r, p, Ap);
    }
}